// GraphSAGE_67353677136083
// MI455X (gfx1250) — compile-verified
//
#include <hip/hip_runtime.h>
#include <math.h>

#define NNODES 40000
#define DFEAT  128
#define NTILES (NNODES / 16)   // 2500 16-node tiles

typedef __attribute__((ext_vector_type(2))) float v2f;
typedef __attribute__((ext_vector_type(8))) float v8f;

// ---------------------------------------------------------------------------
// Degree count: one thread per edge, atomic fp32 add (integer-valued, exact).
// ---------------------------------------------------------------------------
__global__ void count_kernel(const int* __restrict__ dst, float* __restrict__ cnt,
                             int nEdges) {
    int i = blockIdx.x * blockDim.x + threadIdx.x;
    if (i < nEdges) atomicAdd(cnt + dst[i], 1.0f);
}

// cnt[i] = 1 / max(cnt[i], 1)
__global__ void inv_kernel(float* __restrict__ cnt, int n) {
    int i = blockIdx.x * blockDim.x + threadIdx.x;
    if (i < n) cnt[i] = 1.0f / fmaxf(cnt[i], 1.0f);
}

// ---------------------------------------------------------------------------
// Edge scatter: one wave32 per edge, float4 per lane = one 512B feature row.
// Grid-strided (4 edges/wave) with global_prefetch of the next source row.
// agg (20.5MB) and feat (20.5MB) are L2-resident on MI455X (192MB L2), so
// this is L2-atomic-throughput bound, not HBM bound.
// ---------------------------------------------------------------------------
__global__ void scatter_kernel(const float* __restrict__ feat,
                               const int* __restrict__ src,
                               const int* __restrict__ dst,
                               float* __restrict__ agg, int nEdges) {
    const int lane   = threadIdx.x & 31;
    const int wave   = (blockIdx.x * blockDim.x + threadIdx.x) >> 5;
    const int nWaves = (gridDim.x * blockDim.x) >> 5;
    for (int e = wave; e < nEdges; e += nWaves) {
        const int s = src[e];
        const int d = dst[e];
        // Prefetch next iteration's source row (4 x 128B) -> global_prefetch_b8
        const int en = e + nWaves;
        if (en < nEdges && lane < 4) {
            __builtin_prefetch(feat + (size_t)src[en] * DFEAT + lane * 32, 0, 1);
        }
        const float4 v = ((const float4*)(feat + (size_t)s * DFEAT))[lane];
        float* b = agg + (size_t)d * DFEAT + lane * 4;
        atomicAdd(b + 0, v.x);
        atomicAdd(b + 1, v.y);
        atomicAdd(b + 2, v.z);
        atomicAdd(b + 3, v.w);
    }
}

// ---------------------------------------------------------------------------
// SAGE layer: out = L2norm( mean @ Wl^T + bl + feat @ Wr^T ), optional ReLU.
// One wave per 16-node tile, owning all 8 N-tiles (16x128 output) so the
// per-row L2 norm reduces entirely inside the wave.
//
// V_WMMA_F32_16X16X4_F32 fragment layouts (ISA 7.12.2):
//   A 16x4 : lane L holds M=L&15; VGPR0=K(kh), VGPR1=K(kh+1), kh=(L>=16)?2:0
//   B 4x16 : lane L holds N=L&15; VGPR0=K(kh), VGPR1=K(kh+1)   (B[k][n]=W[n][k])
//   C 16x16: VGPR r holds row r (lanes 0-15) / row r+8 (lanes 16-31), N=L&15
// ---------------------------------------------------------------------------
__global__ void __launch_bounds__(256)
sage_layer_kernel(const float* __restrict__ feat, const float* __restrict__ agg,
                  const float* __restrict__ cntinv,
                  const float* __restrict__ Wl, const float* __restrict__ bl,
                  const float* __restrict__ Wr,
                  float* __restrict__ out, int do_relu) {
    const int lane = threadIdx.x & 31;
    const int tile = blockIdx.x * 8 + (threadIdx.x >> 5);
    if (tile >= NTILES) return;               // whole-wave exit: EXEC stays all-1s

    const int m  = lane & 15;                 // node row (A) / out column (B)
    const int kh = (lane >> 4) << 1;          // 0 or 2
    const int nodeBase = tile * 16;

    const float  inv     = cntinv[nodeBase + m];
    const float* aggRow  = agg  + (size_t)(nodeBase + m) * DFEAT;
    const float* featRow = feat + (size_t)(nodeBase + m) * DFEAT;

    v8f acc[8] = {};                          // 8 N-tiles x 8 VGPRs

    for (int k0 = 0; k0 < DFEAT; k0 += 4) {
        v2f aN, aR;                           // neighbor-mean and root fragments
        aN.x = aggRow[k0 + kh]     * inv;
        aN.y = aggRow[k0 + kh + 1] * inv;
        aR.x = featRow[k0 + kh];
        aR.y = featRow[k0 + kh + 1];
#pragma unroll
        for (int t = 0; t < 8; ++t) {
            const size_t wrow = (size_t)(t * 16 + m) * DFEAT + k0 + kh;
            v2f bL, bR;
            bL.x = Wl[wrow];  bL.y = Wl[wrow + 1];
            bR.x = Wr[wrow];  bR.y = Wr[wrow + 1];
            acc[t] = __builtin_amdgcn_wmma_f32_16x16x4_f32(
                false, aN, false, bL, (short)0, acc[t], false, false);
            acc[t] = __builtin_amdgcn_wmma_f32_16x16x4_f32(
                false, aR, false, bR, (short)0, acc[t], false, false);
        }
    }

    // bias (depends only on output column)
#pragma unroll
    for (int t = 0; t < 8; ++t) {
        const float b = bl[t * 16 + m];
#pragma unroll
        for (int r = 0; r < 8; ++r) acc[t][r] += b;
    }

    // per-row sum of squares; reduce across the 16 lanes of each half-wave
    float ss[8];
#pragma unroll
    for (int r = 0; r < 8; ++r) {
        float s = 0.0f;
#pragma unroll
        for (int t = 0; t < 8; ++t) s += acc[t][r] * acc[t][r];
        ss[r] = s;
    }
#pragma unroll
    for (int off = 1; off < 16; off <<= 1) {
#pragma unroll
        for (int r = 0; r < 8; ++r) ss[r] += __shfl_xor(ss[r], off, 32);
    }

    float scl[8];
#pragma unroll
    for (int r = 0; r < 8; ++r)
        scl[r] = 1.0f / fmaxf(sqrtf(ss[r]), 1e-12f);

    const int half = lane >> 4;               // row offset 0 or 8
#pragma unroll
    for (int t = 0; t < 8; ++t) {
#pragma unroll
        for (int r = 0; r < 8; ++r) {
            float v = acc[t][r] * scl[r];
            if (do_relu) v = fmaxf(v, 0.0f);
            out[(size_t)(nodeBase + r + 8 * half) * DFEAT + t * 16 + m] = v;
        }
    }
}

// ---------------------------------------------------------------------------
// Host launcher. Inputs: x, edge_index, Wl1, bl1, Wr1, Wl2, bl2, Wr2.
// Workspace: agg[N*D] | cnt[N] | h[N*D]  (~41.2 MB).
// ---------------------------------------------------------------------------
extern "C" void kernel_launch(void* const* d_in, const int* in_sizes, int n_in,
                              void* d_out, int out_size, void* d_ws, size_t ws_size,
                              hipStream_t stream) {
    const float* x   = (const float*)d_in[0];
    const int*   ei  = (const int*)d_in[1];
    const float* Wl1 = (const float*)d_in[2];
    const float* bl1 = (const float*)d_in[3];
    const float* Wr1 = (const float*)d_in[4];
    const float* Wl2 = (const float*)d_in[5];
    const float* bl2 = (const float*)d_in[6];
    const float* Wr2 = (const float*)d_in[7];

    const int nEdges = in_sizes[1] / 2;
    const int* src = ei;
    const int* dst = ei + nEdges;

    float* agg = (float*)d_ws;                      // NNODES * DFEAT
    float* cnt = agg + (size_t)NNODES * DFEAT;      // NNODES
    float* h   = cnt + NNODES;                      // NNODES * DFEAT

    const size_t aggBytes = (size_t)NNODES * DFEAT * sizeof(float);

    hipMemsetAsync(agg, 0, aggBytes, stream);
    hipMemsetAsync(cnt, 0, (size_t)NNODES * sizeof(float), stream);

    count_kernel<<<(nEdges + 255) / 256, 256, 0, stream>>>(dst, cnt, nEdges);
    inv_kernel<<<(NNODES + 255) / 256, 256, 0, stream>>>(cnt, NNODES);

    const int scatterBlocks = 20000;                // 160k waves -> 4 edges/wave
    const int layerBlocks   = (NTILES + 7) / 8;     // 8 waves/block, 1 tile/wave

    // Layer 1: aggregate x, GEMM + bias + L2norm + ReLU -> h
    scatter_kernel<<<scatterBlocks, 256, 0, stream>>>(x, src, dst, agg, nEdges);
    sage_layer_kernel<<<layerBlocks, 256, 0, stream>>>(x, agg, cnt,
                                                       Wl1, bl1, Wr1, h, 1);

    // Layer 2: aggregate h, GEMM + bias + L2norm -> d_out
    hipMemsetAsync(agg, 0, aggBytes, stream);
    scatter_kernel<<<scatterBlocks, 256, 0, stream>>>(h, src, dst, agg, nEdges);
    sage_layer_kernel<<<layerBlocks, 256, 0, stream>>>(h, agg, cnt,
                                                       Wl2, bl2, Wr2,
                                                       (float*)d_out, 0);
}